// TPSTransformerLayer_51908974739992
// MI455X (gfx1250) — compile-verified
//
#include <hip/hip_runtime.h>
#include <hip/hip_bf16.h>

typedef __attribute__((ext_vector_type(2))) float v2f;
typedef __attribute__((ext_vector_type(8))) float v8f;

#define TPS_H 512
#define TPS_W 512
#define TPS_C 3
#define TPS_NT 4          // 16-wide n-tiles per wave
#define TPS_GROUPS 4      // 4 groups of 8 batches = 32 batches

// ---------------------------------------------------------------------------
// Kernel 1: coeff[b,p,e] = sum_k (src[p,k] + doff[b,p,k]) * L_inv[e, 3+k]
// stored WMMA-A-friendly: wsA[g*320 + e*16 + m], m = (b%8)*2 + p, e in [0,20)
// (e == 19 is zero padding so the K=20 WMMA chain is exact).
// ---------------------------------------------------------------------------
__global__ void tps_coeff_kernel(const float* __restrict__ doff,
                                 const float* __restrict__ L_inv,
                                 const float* __restrict__ src,
                                 float* __restrict__ wsA) {
    int t = blockIdx.x * blockDim.x + threadIdx.x;
    if (t >= TPS_GROUPS * 20 * 16) return;
    int g = t / 320;
    int r = t % 320;
    int e = r / 16;
    int m = r % 16;
    float acc = 0.0f;
    if (e < 19) {
        int b = g * 8 + (m >> 1);
        int p = m & 1;
        #pragma unroll
        for (int k = 0; k < 16; ++k) {
            float d = src[p * 16 + k] + doff[b * 32 + p * 16 + k];
            acc = fmaf(d, L_inv[e * 19 + 3 + k], acc);
        }
    }
    wsA[t] = acc;
}

// ---------------------------------------------------------------------------
// Kernel 2: fused  tp = coeff @ right_mat  (V_WMMA_F32_16X16X4_F32)
//           + bilinear gather/blend, never materializing tp in HBM.
// Wave: batch-group g (8 batches, M = 2*bb+p), 4 tiles of 16 n-columns.
// D layout: VGPR r, lane l -> M = (l<16 ? r : r+8), N = l%16
//   => lane holds (x,y) for 4 batches at one column: c[2q]=x, c[2q+1]=y.
// ---------------------------------------------------------------------------
__global__ void __launch_bounds__(128)
tps_sample_kernel(const float* __restrict__ image,
                  const float* __restrict__ right_mat,
                  const float* __restrict__ wsA,
                  float* __restrict__ out,
                  int Nn) {
    const int lane = threadIdx.x & 31;
    const int g    = threadIdx.x >> 5;      // batch group 0..3
    const int half = lane >> 4;             // lane half selects K pair / M+8
    const int col  = lane & 15;             // N column within tile

    // A matrix (coeff), 5 K-steps of 4, held in registers across all tiles.
    float a0[5], a1[5];
    #pragma unroll
    for (int k = 0; k < 5; ++k) {
        int e0 = 4 * k + (half ? 2 : 0);
        a0[k] = wsA[g * 320 + e0 * 16 + col];
        a1[k] = wsA[g * 320 + (e0 + 1) * 16 + col];
    }

    const int nbase = blockIdx.x * (16 * TPS_NT);

    #pragma unroll 1
    for (int t = 0; t < TPS_NT; ++t) {
        const int n0 = nbase + t * 16;

        v8f c = {0.f, 0.f, 0.f, 0.f, 0.f, 0.f, 0.f, 0.f};
        #pragma unroll
        for (int k = 0; k < 5; ++k) {
            int e0 = 4 * k + (half ? 2 : 0);
            int e1 = e0 + 1;
            // clamp B row index for the e==19 pad (A row there is zero)
            int ec0 = e0 > 18 ? 18 : e0;
            int ec1 = e1 > 18 ? 18 : e1;
            v2f a, b;
            a[0] = a0[k];
            a[1] = a1[k];
            b[0] = right_mat[ec0 * Nn + n0 + col];
            b[1] = right_mat[ec1 * Nn + n0 + col];
            // D = A(16x4) * B(4x16) + C, f32 accumulate
            c = __builtin_amdgcn_wmma_f32_16x16x4_f32(
                    false, a, false, b, (short)0, c, false, false);
        }

        const int n = n0 + col;
        #pragma unroll
        for (int q = 0; q < 4; ++q) {
            const float xr = c[2 * q];       // M even  -> x coord
            const float yr = c[2 * q + 1];   // M odd   -> y coord
            const int b = g * 8 + half * 4 + q;

            float x = 0.5f * (xr + 1.0f) * (float)TPS_W;
            float y = 0.5f * (yr + 1.0f) * (float)TPS_H;
            int x0 = (int)x;                 // trunc toward zero (matches cast)
            int y0 = (int)y;
            int x1 = x0 + 1;
            int y1 = y0 + 1;
            x0 = x0 < 0 ? 0 : (x0 > TPS_W - 1 ? TPS_W - 1 : x0);
            x1 = x1 < 0 ? 0 : (x1 > TPS_W - 1 ? TPS_W - 1 : x1);
            y0 = y0 < 0 ? 0 : (y0 > TPS_H - 1 ? TPS_H - 1 : y0);
            y1 = y1 < 0 ? 0 : (y1 > TPS_H - 1 ? TPS_H - 1 : y1);

            const float x0f = (float)x0, x1f = (float)x1;
            const float y0f = (float)y0, y1f = (float)y1;
            const float wa = (x1f - x) * (y1f - y);
            const float wb = (x1f - x) * (y - y0f);
            const float wc = (x - x0f) * (y1f - y);
            const float wd = (x - x0f) * (y - y0f);

            const float* img = image + b * (TPS_H * TPS_W * TPS_C);
            const float* pa = img + (y0 * TPS_W + x0) * TPS_C;
            const float* pb = img + (y1 * TPS_W + x0) * TPS_C;
            const float* pc = img + (y0 * TPS_W + x1) * TPS_C;
            const float* pd = img + (y1 * TPS_W + x1) * TPS_C;

            float* o = out + ((long long)b * Nn + n) * TPS_C;
            #pragma unroll
            for (int ch = 0; ch < 3; ++ch) {
                o[ch] = wa * pa[ch] + wb * pb[ch] + wc * pc[ch] + wd * pd[ch];
            }
        }
    }
}

// ---------------------------------------------------------------------------
extern "C" void kernel_launch(void* const* d_in, const int* in_sizes, int n_in,
                              void* d_out, int out_size, void* d_ws, size_t ws_size,
                              hipStream_t stream) {
    const float* image = (const float*)d_in[0];   // (32,512,512,3)
    const float* doff  = (const float*)d_in[1];   // (32,32)
    const float* rmat  = (const float*)d_in[2];   // (19,N)
    const float* linv  = (const float*)d_in[3];   // (19,19)
    const float* src   = (const float*)d_in[4];   // (2,16)
    float* out = (float*)d_out;
    float* wsA = (float*)d_ws;                    // 1280 floats

    const int Nn = in_sizes[2] / 19;              // 262144

    tps_coeff_kernel<<<5, 256, 0, stream>>>(doff, linv, src, wsA);

    const int blocks = Nn / (16 * TPS_NT);        // 4096, exact (no partial tiles
                                                  // -> EXEC all-ones for WMMA)
    tps_sample_kernel<<<blocks, 128, 0, stream>>>(image, rmat, wsA, out, Nn);
}